// GraphConvLayer_2405181685924
// MI455X (gfx1250) — compile-verified
//
#include <hip/hip_runtime.h>

// Problem constants (from the reference)
#define B_   16
#define CIN  32
#define T_   128
#define N_   512
#define CO   64
#define KS_  3

// LDS row strides
#define ST   516  // channel-major buffers: [ch][node 0..511] + 4 pad  (bank-conflict-free b64/b128)
#define WRS  34   // W^T buffers:           [j][i 0..31] + 2 pad

typedef __attribute__((ext_vector_type(2))) float v2f;
typedef __attribute__((ext_vector_type(4))) float v4f;
typedef __attribute__((ext_vector_type(8))) float v8f;

__device__ __forceinline__ v8f wmma_f32x4(v2f a, v2f b, v8f c) {
  // 8 args: (neg_a, A, neg_b, B, c_mod, C, reuse_a, reuse_b)
  return __builtin_amdgcn_wmma_f32_16x16x4_f32(
      false, a, false, b, (short)0, c, false, false);
}

__launch_bounds__(512, 1)
__global__ void cheb_graph_conv_kernel(const float* __restrict__ x,
                                       const float* __restrict__ gso,
                                       const float* __restrict__ weight,
                                       const float* __restrict__ bias,
                                       float* __restrict__ out) {
  // Channel-major staging: only the 32 real channels are materialized.
  // B-fragment wants (K=node contiguous per lane, N=ch per lane) -> one ds_load_b64.
  __shared__ float XsT [CIN * ST];        // X  [ch][node]
  __shared__ float Y1sT[CIN * ST];        // Y1 [ch][node]
  __shared__ float Y2sT[CIN * ST];        // Y2 [ch][node]
  __shared__ float WsT [KS_ * CO * WRS];  // W^T [k][j][i]
  __shared__ float Bs  [CO];

  const int tid  = threadIdx.x;
  const int lane = tid & 31;
  const int wave = tid >> 5;        // 0..15
  const int lo   = lane & 15;       // M/N index within a 16-wide tile
  const int hh   = lane >> 4;       // lane half
  const int hi2  = hh << 1;         // K offset (A/B frags): K = hi2 + vgpr
  const int hi8  = hh << 3;         // M offset (C/D tiles): M = r + hi8

  const int bt = blockIdx.x;        // 0..2047
  const int b  = bt >> 7;           // / T_
  const int t  = bt & (T_ - 1);

  // ---------------- Phase A: stage X, W^T, bias into LDS ----------------
  for (int idx = tid; idx < CIN * N_; idx += 512) {
    int c = idx >> 9;               // / N_
    int h = idx & (N_ - 1);
    // x[b][c][t][h]  (coalesced along h; LDS banks consecutive along h)
    XsT[c * ST + h] = x[(((size_t)b * CIN + c) * T_ + t) * N_ + h];
  }
  for (int idx = tid; idx < KS_ * CIN * CO; idx += 512) {
    int j  = idx & (CO - 1);
    int ki = idx >> 6;
    int k  = ki >> 5;               // / CIN
    int i  = ki & (CIN - 1);
    // weight[k][i][j] -> WsT[k][j][i]   (coalesced along j)
    WsT[(k * CO + j) * WRS + i] = weight[((size_t)k * CO + i) * CO + j];
  }
  if (tid < CO) Bs[tid] = bias[tid];
  __syncthreads();

  const int h_base = wave * 32;     // this wave's 32-node output stripe

  // ---------------- Phase B: Y1 = G @ X  (M=node, N=ch, K=node) ----------
  {
    v8f acc[2][2] = {};
    #pragma unroll 2
    for (int k0 = 0; k0 < N_; k0 += 4) {
      // A frags: rows of gso, float2 per lane (8B, L2-resident matrix)
      v2f a0 = *(const v2f*)(gso + (size_t)(h_base +      lo) * N_ + k0 + hi2);
      v2f a1 = *(const v2f*)(gso + (size_t)(h_base + 16 + lo) * N_ + k0 + hi2);
      // B frags: element (K = node, N = ch): contiguous b64 from channel-major LDS
      v2f b0 = *(const v2f*)(XsT + (     lo) * ST + k0 + hi2);
      v2f b1 = *(const v2f*)(XsT + (16 + lo) * ST + k0 + hi2);
      acc[0][0] = wmma_f32x4(a0, b0, acc[0][0]);
      acc[0][1] = wmma_f32x4(a0, b1, acc[0][1]);
      acc[1][0] = wmma_f32x4(a1, b0, acc[1][0]);
      acc[1][1] = wmma_f32x4(a1, b1, acc[1][1]);
    }
    // Stripe write-back: consecutive r = consecutive node -> two b128 per tile
    #pragma unroll
    for (int mt = 0; mt < 2; ++mt)
      #pragma unroll
      for (int nt = 0; nt < 2; ++nt) {
        float* dst = Y1sT + (16 * nt + lo) * ST + h_base + 16 * mt + hi8;
        v4f p0 = {acc[mt][nt][0], acc[mt][nt][1], acc[mt][nt][2], acc[mt][nt][3]};
        v4f p1 = {acc[mt][nt][4], acc[mt][nt][5], acc[mt][nt][6], acc[mt][nt][7]};
        *(v4f*)(dst)     = p0;
        *(v4f*)(dst + 4) = p1;
      }
  }
  __syncthreads();

  // ---------------- Phase C: Y2 = 2*(G @ Y1) - X --------------------------
  {
    v8f acc[2][2] = {};
    #pragma unroll 2
    for (int k0 = 0; k0 < N_; k0 += 4) {
      v2f a0 = *(const v2f*)(gso + (size_t)(h_base +      lo) * N_ + k0 + hi2);
      v2f a1 = *(const v2f*)(gso + (size_t)(h_base + 16 + lo) * N_ + k0 + hi2);
      v2f b0 = *(const v2f*)(Y1sT + (     lo) * ST + k0 + hi2);
      v2f b1 = *(const v2f*)(Y1sT + (16 + lo) * ST + k0 + hi2);
      acc[0][0] = wmma_f32x4(a0, b0, acc[0][0]);
      acc[0][1] = wmma_f32x4(a0, b1, acc[0][1]);
      acc[1][0] = wmma_f32x4(a1, b0, acc[1][0]);
      acc[1][1] = wmma_f32x4(a1, b1, acc[1][1]);
    }
    #pragma unroll
    for (int mt = 0; mt < 2; ++mt)
      #pragma unroll
      for (int nt = 0; nt < 2; ++nt) {
        int off = (16 * nt + lo) * ST + h_base + 16 * mt + hi8;
        v4f x0 = *(const v4f*)(XsT + off);
        v4f x1 = *(const v4f*)(XsT + off + 4);
        v4f y0 = {2.0f * acc[mt][nt][0] - x0.x, 2.0f * acc[mt][nt][1] - x0.y,
                  2.0f * acc[mt][nt][2] - x0.z, 2.0f * acc[mt][nt][3] - x0.w};
        v4f y1 = {2.0f * acc[mt][nt][4] - x1.x, 2.0f * acc[mt][nt][5] - x1.y,
                  2.0f * acc[mt][nt][6] - x1.z, 2.0f * acc[mt][nt][7] - x1.w};
        *(v4f*)(Y2sT + off)     = y0;
        *(v4f*)(Y2sT + off + 4) = y1;
      }
  }
  __syncthreads();

  // ------- Phase D: OUT^T = sum_k W_k^T @ xs_k^T  (M=ch j, N=node h) -----
  // Computing the transpose makes the D-tile lanes map to 16 consecutive
  // node addresses -> coalesced 64B stores into out[B][CO][T][N].
  {
    v8f acc[4][2] = {};   // [j-tile 0..3][h-tile 0..1]

    auto accumulate = [&](const float* __restrict__ xb,
                          const float* __restrict__ wb) {
      #pragma unroll
      for (int i0 = 0; i0 < CIN; i0 += 4) {
        v2f a[4], bf[2];
        #pragma unroll
        for (int jt = 0; jt < 4; ++jt)   // A = W^T tile: (M=j, K=i) contiguous
          a[jt] = *(const v2f*)(wb + (16 * jt + lo) * WRS + i0 + hi2);
        #pragma unroll
        for (int ht = 0; ht < 2; ++ht) { // B = xs^T tile: (K=i, N=h) cross-row
          int h = h_base + 16 * ht + lo;
          v2f bb;
          bb.x = xb[(i0 + hi2    ) * ST + h];
          bb.y = xb[(i0 + hi2 + 1) * ST + h];
          bf[ht] = bb;
        }
        #pragma unroll
        for (int jt = 0; jt < 4; ++jt)
          #pragma unroll
          for (int ht = 0; ht < 2; ++ht)
            acc[jt][ht] = wmma_f32x4(a[jt], bf[ht], acc[jt][ht]);
      }
    };
    accumulate(XsT,  WsT);                 // k = 0
    accumulate(Y1sT, WsT +     CO * WRS);  // k = 1
    accumulate(Y2sT, WsT + 2 * CO * WRS);  // k = 2

    // Epilogue: bias + residual (x_in pad: only j<32 contributes), store.
    #pragma unroll
    for (int jt = 0; jt < 4; ++jt) {
      #pragma unroll
      for (int ht = 0; ht < 2; ++ht) {
        int h = h_base + 16 * ht + lo;
        #pragma unroll
        for (int r = 0; r < 8; ++r) {
          int j = 16 * jt + hi8 + r;
          float v = acc[jt][ht][r] + Bs[j];
          if (j < CIN) v += XsT[j * ST + h];
          out[(((size_t)b * CO + j) * T_ + t) * N_ + h] = v;
        }
      }
    }
  }
}

extern "C" void kernel_launch(void* const* d_in, const int* in_sizes, int n_in,
                              void* d_out, int out_size, void* d_ws, size_t ws_size,
                              hipStream_t stream) {
  const float* x      = (const float*)d_in[0];
  const float* gso    = (const float*)d_in[1];
  const float* weight = (const float*)d_in[2];
  const float* bias   = (const float*)d_in[3];
  float* out          = (float*)d_out;

  cheb_graph_conv_kernel<<<dim3(B_ * T_), dim3(512), 0, stream>>>(
      x, gso, weight, bias, out);
}